// Router_506806141650
// MI455X (gfx1250) — compile-verified
//
#include <hip/hip_runtime.h>

typedef float v2f __attribute__((ext_vector_type(2)));
typedef float v8f __attribute__((ext_vector_type(8)));

#define INPUT_DIM 2048
#define NUM_EXPERTS 8
#define TOKENS_PER_WAVE 16
#define WAVES_PER_BLOCK 8
#define TOKENS_PER_BLOCK (TOKENS_PER_WAVE * WAVES_PER_BLOCK)

__global__ __launch_bounds__(256) void router_wmma_kernel(
    const float* __restrict__ x,   // [nTokens, INPUT_DIM]
    const float* __restrict__ W,   // [NUM_EXPERTS, INPUT_DIM]
    const float* __restrict__ bias,// [NUM_EXPERTS]
    float* __restrict__ outW,      // [nTokens, 2]
    int* __restrict__ outI)        // [nTokens, 2]
{
    __shared__ float lds[WAVES_PER_BLOCK][TOKENS_PER_WAVE][NUM_EXPERTS];

    const int lane = threadIdx.x & 31;
    const int wave = threadIdx.x >> 5;
    const int tokenBase = blockIdx.x * TOKENS_PER_BLOCK + wave * TOKENS_PER_WAVE;

    const int lo = lane & 15;   // A: row M / B: col N
    const int hi = lane >> 4;   // K sub-group: K += 2*hi

    // A: x[tokenBase+lo][k + 2*hi + {0,1}] -> contiguous 8B per lane
    const float* arow = x + (size_t)(tokenBase + lo) * INPUT_DIM + 2 * hi;
    // B: W^T[k][lo] = W[lo][k]; zero columns lo >= 8 (branchless, EXEC stays all-1s)
    const float* brow = W + (size_t)(lo & (NUM_EXPERTS - 1)) * INPUT_DIM + 2 * hi;
    const float bsel = (lo < NUM_EXPERTS) ? 1.0f : 0.0f;

    v8f acc = {};
#pragma unroll 4
    for (int k = 0; k < INPUT_DIM; k += 4) {
        v2f a  = *(const v2f*)(arow + k);
        v2f bb = *(const v2f*)(brow + k);
        bb.x *= bsel;
        bb.y *= bsel;
        // D = A(16x4) * B(4x16) + C  — f32 WMMA, exact reference numerics
        acc = __builtin_amdgcn_wmma_f32_16x16x4_f32(
            /*neg_a=*/false, a, /*neg_b=*/false, bb,
            /*c_mod=*/(short)0, acc, /*reuse_a=*/false, /*reuse_b=*/false);
    }

    // C/D layout: VGPR j, lane l -> token = j + 8*(l>=16), expert = l&15.
    // Spill valid experts (lo < 8) to LDS to transpose into per-token rows.
    if (lo < NUM_EXPERTS) {
#pragma unroll
        for (int j = 0; j < 8; ++j)
            lds[wave][j + 8 * hi][lo] = acc[j];
    }
    __syncthreads();

    if (lane < TOKENS_PER_WAVE) {
        const int tok = tokenBase + lane;

        float lg[NUM_EXPERTS];
        float m = -3.402823466e+38f;
#pragma unroll
        for (int e = 0; e < NUM_EXPERTS; ++e) {
            lg[e] = lds[wave][lane][e] + bias[e];  // bias: uniform scalar loads
            m = fmaxf(m, lg[e]);
        }
        float s = 0.0f;
#pragma unroll
        for (int e = 0; e < NUM_EXPERTS; ++e) {
            lg[e] = __expf(lg[e] - m);
            s += lg[e];
        }
        const float inv = 1.0f / s;

        // top-1 (strict > : lowest index wins ties, matches jax.lax.top_k)
        float best = -1.0f; int bi = 0;
#pragma unroll
        for (int e = 0; e < NUM_EXPERTS; ++e) {
            if (lg[e] > best) { best = lg[e]; bi = e; }
        }
        // top-2
        float second = -1.0f; int si = 0;
#pragma unroll
        for (int e = 0; e < NUM_EXPERTS; ++e) {
            if (e != bi && lg[e] > second) { second = lg[e]; si = e; }
        }

        outW[2 * tok + 0] = best * inv;
        outW[2 * tok + 1] = second * inv;
        outI[2 * tok + 0] = bi;
        outI[2 * tok + 1] = si;
    }
}

extern "C" void kernel_launch(void* const* d_in, const int* in_sizes, int n_in,
                              void* d_out, int out_size, void* d_ws, size_t ws_size,
                              hipStream_t stream) {
    const float* x = (const float*)d_in[0];
    const float* W = (const float*)d_in[1];
    const float* b = (const float*)d_in[2];
    // d_in[3] is k (== 2), compile-time constant here.

    const int nTokens = in_sizes[0] / INPUT_DIM;  // 32768

    float* outW = (float*)d_out;                         // [nTokens, 2] f32
    int*   outI = (int*)((float*)d_out + (size_t)nTokens * 2);  // [nTokens, 2] i32

    const int blocks = nTokens / TOKENS_PER_BLOCK;  // 256
    router_wmma_kernel<<<blocks, 256, 0, stream>>>(x, W, b, outW, outI);
}